// RNN_352187319062
// MI455X (gfx1250) — compile-verified
//
#include <hip/hip_runtime.h>
#include <cstdint>

#define T_STEPS 512
#define BATCH   64
#define IN_DIM  1024
#define HID     1024

typedef float v2f  __attribute__((ext_vector_type(2)));
typedef float v8f  __attribute__((ext_vector_type(8)));
typedef unsigned int u32x4 __attribute__((ext_vector_type(4)));
typedef int   i32x8 __attribute__((ext_vector_type(8)));
typedef int   i32x4 __attribute__((ext_vector_type(4)));

// D = A(16x4, f32) * B(4x16, f32) + C(16x16, f32)
__device__ __forceinline__ v8f wmma_f32_16x16x4(v2f a, v2f b, v8f c) {
  return __builtin_amdgcn_wmma_f32_16x16x4_f32(false, a, false, b, (short)0, c,
                                               false, false);
}

// Low 32 bits of a generic LDS pointer = offset within the workgroup's LDS.
__device__ __forceinline__ unsigned lds_off(const void* p) {
  return (unsigned)(unsigned long long)(uintptr_t)p;
}

// ---------------------------------------------------------------------------
// TDM: DMA a [rows x 32] f32 tile (tensor row stride 1024 f32) into LDS with
// a 36-dword padded row stride (pad 4 dwords after every 32 dwords stored).
// D# packing per CDNA5 ISA.
// ---------------------------------------------------------------------------
__device__ __forceinline__ void tdm_load_tile32(unsigned lds_addr,
                                                const float* gptr,
                                                int rows) {
  unsigned long long ga = (unsigned long long)(uintptr_t)gptr;
  u32x4 g0;
  g0[0] = 1u;                                     // count=1, user mode
  g0[1] = lds_addr;                               // lds_addr
  g0[2] = (unsigned)ga;                           // global_addr[31:0]
  g0[3] = (unsigned)((ga >> 32) & 0x01FFFFFFu)    // global_addr[56:32]
          | (2u << 30);                           // type=2 (image)
  i32x8 g1;
  g1[0] = (2 << 16)                               // data_size = 4B
        | (1 << 20)                               // pad_enable
        | (4 << 22)                               // pad_interval: 32 dwords
        | (3 << 25);                              // pad_amount: 4 dwords
  g1[1] = (IN_DIM & 0xFFFF) << 16;                // tensor_dim0 = 1024 (lo16)
  g1[2] = (rows & 0xFFFF) << 16;                  // dim0 hi=0 | tensor_dim1 lo16
  g1[3] = (32 << 16);                             // dim1 hi=0 | tile_dim0 = 32
  g1[4] = rows & 0xFFFF;                          // tile_dim1 = rows, tile_dim2=0
  g1[5] = IN_DIM;                                 // tensor_dim0_stride lo32
  g1[6] = 0;                                      // stride hi | dim1_stride lo
  g1[7] = 0;
  i32x4 z4 = {0, 0, 0, 0};                        // <=2-D: groups 2/3 unused
  i32x8 z8 = {0, 0, 0, 0, 0, 0, 0, 0};
  __builtin_amdgcn_tensor_load_to_lds(g0, g1, z4, z4, z8, 0);
}

// ---------------------------------------------------------------------------
// Phase 1: x_proj[r, n] = sum_k input[r,k] * W_ih[n,k] + b_ih[n] + b_hh[n]
// M = T*B = 32768, N = H = 1024, K = I = 1024
// Block tile 128(M) x 64(N), 8 waves (4x2), wave tile 32x32 (2x2 WMMA tiles).
// TDM double-buffered LDS staging.
// ---------------------------------------------------------------------------
__global__ __launch_bounds__(256) void xproj_gemm(
    const float* __restrict__ inp,   // [T*B, I] row-major
    const float* __restrict__ Wih,   // [H, I]   row-major
    const float* __restrict__ bih,   // [H]
    const float* __restrict__ bhh,   // [H]
    float* __restrict__ out)         // [T*B, H]
{
  __shared__ float As[2][128][36];   // padded stride 36: conflict-free frags
  __shared__ float Bs[2][64][36];

  const int tid  = threadIdx.x;
  const int lane = tid & 31;
  const int wave = tid >> 5;
  const int half = lane >> 4;        // 0: K sub 0/1, 1: K sub 2/3
  const int ln   = lane & 15;
  const int wm   = wave >> 1;        // 0..3 -> M offset 32*wm
  const int wn   = wave & 1;         // 0..1 -> N offset 32*wn
  const int r0   = blockIdx.y * 128;
  const int n0   = blockIdx.x * 64;

  v8f acc00 = {}, acc01 = {}, acc10 = {}, acc11 = {};

  // Preload chunk 0 via TDM (wave 0 only; TENSORcnt is per-wave).
  if (wave == 0) {
    tdm_load_tile32(lds_off(&As[0][0][0]), inp + (size_t)r0 * IN_DIM, 128);
    tdm_load_tile32(lds_off(&Bs[0][0][0]), Wih + (size_t)n0 * IN_DIM, 64);
    __builtin_amdgcn_s_wait_tensorcnt(0);
  }
  __syncthreads();

  int p = 0;
  for (int k0 = 0; k0 < IN_DIM; k0 += 32) {
    // Kick off DMA of the next K chunk into the other buffer.
    if (k0 + 32 < IN_DIM && wave == 0) {
      tdm_load_tile32(lds_off(&As[p ^ 1][0][0]),
                      inp + (size_t)r0 * IN_DIM + k0 + 32, 128);
      tdm_load_tile32(lds_off(&Bs[p ^ 1][0][0]),
                      Wih + (size_t)n0 * IN_DIM + k0 + 32, 64);
    }

#pragma unroll
    for (int kk = 0; kk < 32; kk += 4) {
      const int kb = kk + 2 * half;
      v2f a0, a1, b0, b1;
      a0[0] = As[p][wm * 32 + ln][kb];        a0[1] = As[p][wm * 32 + ln][kb + 1];
      a1[0] = As[p][wm * 32 + 16 + ln][kb];   a1[1] = As[p][wm * 32 + 16 + ln][kb + 1];
      b0[0] = Bs[p][wn * 32 + ln][kb];        b0[1] = Bs[p][wn * 32 + ln][kb + 1];
      b1[0] = Bs[p][wn * 32 + 16 + ln][kb];   b1[1] = Bs[p][wn * 32 + 16 + ln][kb + 1];
      acc00 = wmma_f32_16x16x4(a0, b0, acc00);
      acc01 = wmma_f32_16x16x4(a0, b1, acc01);
      acc10 = wmma_f32_16x16x4(a1, b0, acc10);
      acc11 = wmma_f32_16x16x4(a1, b1, acc11);
    }

    if (wave == 0) __builtin_amdgcn_s_wait_tensorcnt(0);
    __syncthreads();
    p ^= 1;
  }

  const int col0 = n0 + wn * 32 + ln;
  const int col1 = col0 + 16;
  const float bias0 = bih[col0] + bhh[col0];
  const float bias1 = bih[col1] + bhh[col1];
#pragma unroll
  for (int g = 0; g < 8; ++g) {
    int rowA = r0 + wm * 32 + g + (half ? 8 : 0);
    int rowB = rowA + 16;
    out[(size_t)rowA * HID + col0] = acc00[g] + bias0;
    out[(size_t)rowA * HID + col1] = acc01[g] + bias1;
    out[(size_t)rowB * HID + col0] = acc10[g] + bias0;
    out[(size_t)rowB * HID + col1] = acc11[g] + bias1;
  }
}

// ---------------------------------------------------------------------------
// Phase 2: one recurrent step (in place):
//   slice[b, n] = tanh(slice[b, n] + sum_k hprev[b,k] * W_hh[n,k])
// M = B = 64, N tile = 64 per block (grid 16), K = 1024.
// 8 waves (2x4): wave tile 32(M) x 16(N), TDM double-buffered staging.
// ---------------------------------------------------------------------------
__global__ __launch_bounds__(256) void rnn_step(
    const float* __restrict__ hprev, // [B, H] = h_{t-1}
    const float* __restrict__ Whh,   // [H, H]
    float* __restrict__ slice)       // [B, H]: in = x_proj_t, out = h_t
{
  __shared__ float As[2][64][36];    // h_prev tile
  __shared__ float Bs[2][64][36];    // W_hh tile

  const int tid  = threadIdx.x;
  const int lane = tid & 31;
  const int wave = tid >> 5;
  const int half = lane >> 4;
  const int ln   = lane & 15;
  const int wm   = wave >> 2;        // 0..1 -> M offset 32*wm
  const int wn   = wave & 3;         // 0..3 -> N offset 16*wn
  const int n0   = blockIdx.x * 64;

  v8f acc0 = {}, acc1 = {};

  if (wave == 0) {
    tdm_load_tile32(lds_off(&As[0][0][0]), hprev, 64);
    tdm_load_tile32(lds_off(&Bs[0][0][0]), Whh + (size_t)n0 * HID, 64);
    __builtin_amdgcn_s_wait_tensorcnt(0);
  }
  __syncthreads();

  int p = 0;
  for (int k0 = 0; k0 < HID; k0 += 32) {
    if (k0 + 32 < HID && wave == 0) {
      tdm_load_tile32(lds_off(&As[p ^ 1][0][0]), hprev + k0 + 32, 64);
      tdm_load_tile32(lds_off(&Bs[p ^ 1][0][0]),
                      Whh + (size_t)n0 * HID + k0 + 32, 64);
    }

#pragma unroll
    for (int kk = 0; kk < 32; kk += 4) {
      const int kb = kk + 2 * half;
      v2f a0, a1, b0;
      a0[0] = As[p][wm * 32 + ln][kb];        a0[1] = As[p][wm * 32 + ln][kb + 1];
      a1[0] = As[p][wm * 32 + 16 + ln][kb];   a1[1] = As[p][wm * 32 + 16 + ln][kb + 1];
      b0[0] = Bs[p][wn * 16 + ln][kb];        b0[1] = Bs[p][wn * 16 + ln][kb + 1];
      acc0 = wmma_f32_16x16x4(a0, b0, acc0);
      acc1 = wmma_f32_16x16x4(a1, b0, acc1);
    }

    if (wave == 0) __builtin_amdgcn_s_wait_tensorcnt(0);
    __syncthreads();
    p ^= 1;
  }

  const int col = n0 + wn * 16 + ln;
#pragma unroll
  for (int g = 0; g < 8; ++g) {
    int rowA = wm * 32 + g + (half ? 8 : 0);
    int rowB = rowA + 16;
    size_t iA = (size_t)rowA * HID + col;
    size_t iB = (size_t)rowB * HID + col;
    slice[iA] = tanhf(acc0[g] + slice[iA]);
    slice[iB] = tanhf(acc1[g] + slice[iB]);
  }
}

// h_last = output[T-1]
__global__ __launch_bounds__(256) void copy_h_last(const float* __restrict__ src,
                                                   float* __restrict__ dst,
                                                   int n) {
  int i = blockIdx.x * blockDim.x + threadIdx.x;
  if (i < n) dst[i] = src[i];
}

extern "C" void kernel_launch(void* const* d_in, const int* in_sizes, int n_in,
                              void* d_out, int out_size, void* d_ws, size_t ws_size,
                              hipStream_t stream) {
  (void)in_sizes; (void)n_in; (void)out_size; (void)d_ws; (void)ws_size;

  const float* input  = (const float*)d_in[0];  // [T, B, I]
  const float* hidden = (const float*)d_in[1];  // [B, H]
  const float* Wih    = (const float*)d_in[2];  // [H, I]
  const float* Whh    = (const float*)d_in[3];  // [H, H]
  const float* bih    = (const float*)d_in[4];  // [H]
  const float* bhh    = (const float*)d_in[5];  // [H]
  float* out = (float*)d_out;                   // [T*B*H] output ++ [B*H] h_last

  // Phase 1: x_proj for all timesteps, written straight into the output buffer.
  dim3 g1(HID / 64, (T_STEPS * BATCH) / 128);
  xproj_gemm<<<g1, 256, 0, stream>>>(input, Wih, bih, bhh, out);

  // Phase 2: sequential scan; output[t-1] already holds h_{t-1}.
  for (int t = 0; t < T_STEPS; ++t) {
    const float* hp = (t == 0) ? hidden : out + (size_t)(t - 1) * BATCH * HID;
    rnn_step<<<HID / 64, 256, 0, stream>>>(hp, Whh,
                                           out + (size_t)t * BATCH * HID);
  }

  // h_last
  copy_h_last<<<(BATCH * HID) / 256, 256, 0, stream>>>(
      out + (size_t)(T_STEPS - 1) * BATCH * HID,
      out + (size_t)T_STEPS * BATCH * HID, BATCH * HID);
}